// SpatialBlock_61022895341615
// MI455X (gfx1250) — compile-verified
//
#include <hip/hip_runtime.h>
#include <hip/hip_bf16.h>

// Problem constants (from reference): x[B,C,H,W] fp32, 7x7 conv 2->1, hsigmoid gate.
#define Bn 16
#define Cn 256
#define Hn 128
#define Wn 128
#define HWn (Hn * Wn)        // 16384
#define PADk 3               // 7x7, pad 3

typedef float v2f __attribute__((ext_vector_type(2)));
typedef float v4f __attribute__((ext_vector_type(4)));
typedef float v8f __attribute__((ext_vector_type(8)));

// ---------------------------------------------------------------------------
// Kernel 1: channel-wise max & mean.  Each thread owns 4 consecutive pixels
// (float4), loops over the 256 channels with fully coalesced 16B loads.
// Streams all 256 MB of x exactly once.
// ---------------------------------------------------------------------------
__global__ void sb_reduce_kernel(const float* __restrict__ x,
                                 float* __restrict__ maxP,
                                 float* __restrict__ avgP) {
    int t = blockIdx.x * blockDim.x + threadIdx.x;     // 65536 threads total
    int e = t * 4;                                     // pixel index across all images
    int b = e / HWn;
    int p = e - b * HWn;
    const float* base = x + (size_t)b * Cn * HWn + p;

    v4f mx = { -3.402823466e+38f, -3.402823466e+38f,
               -3.402823466e+38f, -3.402823466e+38f };
    v4f sm = { 0.f, 0.f, 0.f, 0.f };

#pragma unroll 8
    for (int c = 0; c < Cn; ++c) {
        v4f v = *(const v4f*)(base + (size_t)c * HWn);
        mx.x = fmaxf(mx.x, v.x);
        mx.y = fmaxf(mx.y, v.y);
        mx.z = fmaxf(mx.z, v.z);
        mx.w = fmaxf(mx.w, v.w);
        sm.x += v.x; sm.y += v.y; sm.z += v.z; sm.w += v.w;
    }

    const float inv = 1.0f / (float)Cn;
    v4f av = { sm.x * inv, sm.y * inv, sm.z * inv, sm.w * inv };
    *(v4f*)(maxP + e) = mx;
    *(v4f*)(avgP + e) = av;
}

// ---------------------------------------------------------------------------
// Kernel 2: 7x7 conv (2ch -> 1ch) + bias + hard-sigmoid via V_WMMA_F32_16X16X4_F32.
// One wave32 computes 16 consecutive output pixels (one M=16 tile in a row).
// Implicit GEMM: K = im2col(ci,kh,kw) = 98, zero-padded to 100 => 25 WMMA steps.
// B matrix = weight vector broadcast into all 16 N columns, so every D column
// holds the conv result; lanes 0 and 16 extract the N=0 column.
//
// ISA lane layouts (CDNA5 ISA):
//   A 16x4 f32 : lane l -> M = l&15, {VGPR0,VGPR1} = K = 4s + 2*(l>>4) + {0,1}
//   B 4x16 f32 : lane l -> N = l&15, same K-pair mapping (columns uniform here)
//   D 16x16 f32: VGPR r, lane l -> M = r + 8*(l>>4), N = l&15
// ---------------------------------------------------------------------------
__global__ void sb_conv_wmma_kernel(const float* __restrict__ maxP,
                                    const float* __restrict__ avgP,
                                    const float* __restrict__ wts,   // [1,2,7,7] flat: ci*49+kh*7+kw
                                    const float* __restrict__ bias,  // [1]
                                    float* __restrict__ gate) {
    int tid  = blockIdx.x * blockDim.x + threadIdx.x;
    int wave = tid >> 5;
    int lane = tid & 31;
    int m    = lane & 15;       // M row / N col owned by this lane
    int half = lane >> 4;       // 0: K pair {0,1}; 1: K pair {2,3} within a slice

    int P  = wave * 16;         // first pixel of this wave's tile
    int b  = P / HWn;
    int p  = P - b * HWn;
    int h  = p / Wn;
    int w0 = p - h * Wn;        // multiple of 16, tile stays inside one row

    const float* mplane = maxP + (size_t)b * HWn;
    const float* aplane = avgP + (size_t)b * HWn;

    v8f acc = { 0.f, 0.f, 0.f, 0.f, 0.f, 0.f, 0.f, 0.f };

    for (int s = 0; s < 25; ++s) {
        v2f a, wv;
#pragma unroll
        for (int j = 0; j < 2; ++j) {
            int   kk = 4 * s + 2 * half + j;
            float av = 0.f;
            float wf = 0.f;
            if (kk < 98) {
                int ci = kk / 49;
                int r  = kk - ci * 49;
                int kh = r / 7;
                int kw = r - kh * 7;
                int hin = h + kh - PADk;
                int win = w0 + m + kw - PADk;
                bool ok = (hin >= 0) & (hin < Hn) & (win >= 0) & (win < Wn);
                int hc = hin < 0 ? 0 : (hin > Hn - 1 ? Hn - 1 : hin);
                int wc = win < 0 ? 0 : (win > Wn - 1 ? Wn - 1 : win);
                const float* pl = ci ? aplane : mplane;
                float v = pl[hc * Wn + wc];
                av = ok ? v : 0.f;
                wf = wts[kk];
            }
            if (j == 0) { a.x = av; wv.x = wf; }
            else        { a.y = av; wv.y = wf; }
        }
        // D = A(16x4) * B(4x16) + C, fp32 matrix core
        acc = __builtin_amdgcn_wmma_f32_16x16x4_f32(
            /*neg_a=*/false, a, /*neg_b=*/false, wv,
            /*c_mod=*/(short)0, acc, /*reuse_a=*/false, /*reuse_b=*/false);
    }

    if (m == 0) {   // lanes 0 and 16 own the N=0 column: M = r + 8*half
        float bs = bias[0];
        float* g = gate + (size_t)b * HWn + h * Wn + w0 + 8 * half;
#pragma unroll
        for (int r = 0; r < 8; ++r) {
            float v = acc[r] + bs;
            v = fminf(fmaxf(v + 3.0f, 0.0f), 6.0f) * (1.0f / 6.0f);  // hsigmoid
            g[r] = v;
        }
    }
}

// ---------------------------------------------------------------------------
// Kernel 3: out = x * gate (gate broadcast over the channel dim).
// float4 loads/stores; the 1 MB gate plane is L2-resident so its 256x re-read
// costs nothing at HBM level.  Streams 256 MB in + 256 MB out.
// ---------------------------------------------------------------------------
__global__ void sb_apply_kernel(const float* __restrict__ x,
                                const float* __restrict__ gate,
                                float* __restrict__ out) {
    size_t i = (size_t)blockIdx.x * blockDim.x + threadIdx.x;  // one float4 each
    size_t e = i * 4;
    int plane = (int)(e >> 14);          // b*C + c   (HW = 16384 = 2^14)
    int b     = plane >> 8;              // C = 256 = 2^8
    int pp    = (int)(e & (HWn - 1));

    v4f g  = *(const v4f*)(gate + (size_t)b * HWn + pp);
    v4f xv = *(const v4f*)(x + e);
    v4f o  = { xv.x * g.x, xv.y * g.y, xv.z * g.z, xv.w * g.w };
    *(v4f*)(out + e) = o;
}

// ---------------------------------------------------------------------------
extern "C" void kernel_launch(void* const* d_in, const int* in_sizes, int n_in,
                              void* d_out, int out_size, void* d_ws, size_t ws_size,
                              hipStream_t stream) {
    (void)in_sizes; (void)n_in; (void)out_size; (void)ws_size;

    const float* x      = (const float*)d_in[0];   // [16,256,128,128]
    const float* conv_w = (const float*)d_in[1];   // [1,2,7,7] = 98 floats
    const float* conv_b = (const float*)d_in[2];   // [1]
    float*       out    = (float*)d_out;

    // Workspace layout: max plane | avg plane | gate plane  (1 MB each, 3 MB total)
    float* maxP = (float*)d_ws;
    float* avgP = maxP + (size_t)Bn * HWn;
    float* gate = avgP + (size_t)Bn * HWn;

    // Pass 1: channel max/mean.  B*HW/4 = 65536 threads.
    sb_reduce_kernel<<<256, 256, 0, stream>>>(x, maxP, avgP);

    // Pass 2: WMMA conv + hsigmoid.  B*HW/16 = 16384 waves = 524288 threads.
    sb_conv_wmma_kernel<<<2048, 256, 0, stream>>>(maxP, avgP, conv_w, conv_b, gate);

    // Pass 3: broadcast multiply.  B*C*HW/4 = 16,777,216 float4 groups.
    sb_apply_kernel<<<65536, 256, 0, stream>>>(x, gate, out);
}